// SGCNEncoder_81630148428374
// MI455X (gfx1250) — compile-verified
//
#include <hip/hip_runtime.h>

#define NN 100000
#define EE 1600000
#define TILES (NN / 16)   // 6250, exact

typedef __attribute__((ext_vector_type(2))) float v2f;
typedef __attribute__((ext_vector_type(8))) float v8f;

__device__ __forceinline__ void atomAddF(float* p, float v) {
  __hip_atomic_fetch_add(p, v, __ATOMIC_RELAXED, __HIP_MEMORY_SCOPE_AGENT);
}

// One K-chain of V_WMMA_F32_16X16X4_F32.
// Arow:  per-lane A row base (row = lane&15), elements picked at k0 + 2*khalf.
// Wcol:  W + (output column); W is row-major KxN with N=32, so element [k] = Wcol[k*32].
// ascale: fused mean-normalization factor applied to A fragments.
template <int K>
__device__ __forceinline__ v8f wmma_mac(v8f c, const float* __restrict__ Arow,
                                        float ascale, const float* __restrict__ Wcol,
                                        int khalf) {
#pragma unroll
  for (int k0 = 0; k0 < K; k0 += 4) {
    const int k = k0 + khalf * 2;
    v2f a = *(const v2f*)(Arow + k);
    v2f b;
    b.x = Wcol[k * 32];
    b.y = Wcol[(k + 1) * 32];
    a.x *= ascale;
    a.y *= ascale;
    c = __builtin_amdgcn_wmma_f32_16x16x4_f32(false, a, false, b, (short)0, c,
                                              false, false);
  }
  return c;
}

// -------- Edge pass 1: one wave32 per edge; gather x[src] (64 f32), scatter-add
// into sum_p/sum_n[dst] by edge sign; count pos/neg in-degree.
__global__ void edge_pass1(const long long* __restrict__ ei,
                           const float* __restrict__ ew,
                           const float* __restrict__ x,
                           float* __restrict__ sum_p, float* __restrict__ sum_n,
                           float* __restrict__ cnt_p, float* __restrict__ cnt_n) {
  const int e    = (blockIdx.x * blockDim.x + threadIdx.x) >> 5;
  const int lane = threadIdx.x & 31;
  if (e >= EE) return;
  const float w = ew[e];
  if (w == 0.0f) return;
  const long long src = ei[e];
  const long long dst = ei[(long long)EE + e];
  const v2f v = *(const v2f*)(x + src * 64 + lane * 2);
  float* s = ((w > 0.0f) ? sum_p : sum_n) + dst * 64 + lane * 2;
  atomAddF(s, v.x);
  atomAddF(s + 1, v.y);
  if (lane == 0) atomAddF(((w > 0.0f) ? cnt_p : cnt_n) + dst, 1.0f);
}

// -------- Layer 1: per 16-node tile (one wave):
// hp = relu(mean_p(x) @ w1_pos_l + x @ w1_pos_r + b1_pos_r)   (16x32)
// hn = relu(mean_n(x) @ w1_neg_l + x @ w1_neg_r + b1_neg_r)   (16x32)
__global__ void gemm_layer1(const float* __restrict__ x,
                            const float* __restrict__ sum_p,
                            const float* __restrict__ sum_n,
                            const float* __restrict__ cnt_p,
                            const float* __restrict__ cnt_n,
                            const float* __restrict__ w1pl,
                            const float* __restrict__ w1pr,
                            const float* __restrict__ b1p,
                            const float* __restrict__ w1nl,
                            const float* __restrict__ w1nr,
                            const float* __restrict__ b1n,
                            float* __restrict__ hp, float* __restrict__ hn) {
  const int lane  = threadIdx.x & 31;
  const int tile  = blockIdx.x * (blockDim.x >> 5) + (threadIdx.x >> 5);
  if (tile >= TILES) return;          // uniform per wave -> EXEC stays all-ones
  const int col   = lane & 15;
  const int khalf = lane >> 4;
  const int nodeA = tile * 16 + col;  // A-fragment row for this lane

  const float rp = 1.0f / fmaxf(cnt_p[nodeA], 1.0f);
  const float rn = 1.0f / fmaxf(cnt_n[nodeA], 1.0f);
  const float* Ap = sum_p + (size_t)nodeA * 64;
  const float* An = sum_n + (size_t)nodeA * 64;
  const float* Ax = x + (size_t)nodeA * 64;
  const int rowBase = tile * 16 + khalf * 8;

#pragma unroll
  for (int cb = 0; cb < 32; cb += 16) {
    // positive branch
    v8f c;
    {
      const float bv = b1p[cb + col];
#pragma unroll
      for (int r = 0; r < 8; ++r) c[r] = bv;
    }
    c = wmma_mac<64>(c, Ap, rp, w1pl + cb + col, khalf);
    c = wmma_mac<64>(c, Ax, 1.0f, w1pr + cb + col, khalf);
#pragma unroll
    for (int r = 0; r < 8; ++r)
      hp[(size_t)(rowBase + r) * 32 + cb + col] = fmaxf(c[r], 0.0f);

    // negative branch
    v8f d;
    {
      const float bv = b1n[cb + col];
#pragma unroll
      for (int r = 0; r < 8; ++r) d[r] = bv;
    }
    d = wmma_mac<64>(d, An, rn, w1nl + cb + col, khalf);
    d = wmma_mac<64>(d, Ax, 1.0f, w1nr + cb + col, khalf);
#pragma unroll
    for (int r = 0; r < 8; ++r)
      hn[(size_t)(rowBase + r) * 32 + cb + col] = fmaxf(d[r], 0.0f);
  }
}

// -------- Edge pass 2: one wave32 per edge; lanes cover the 32 features.
// pos edge: s1[dst]+=hp[src], s3[dst]+=hn[src]
// neg edge: s2[dst]+=hn[src], s4[dst]+=hp[src]
__global__ void edge_pass2(const long long* __restrict__ ei,
                           const float* __restrict__ ew,
                           const float* __restrict__ hp,
                           const float* __restrict__ hn,
                           float* __restrict__ s1, float* __restrict__ s2,
                           float* __restrict__ s3, float* __restrict__ s4) {
  const int e    = (blockIdx.x * blockDim.x + threadIdx.x) >> 5;
  const int lane = threadIdx.x & 31;
  if (e >= EE) return;
  const float w = ew[e];
  if (w == 0.0f) return;
  const long long src = ei[e];
  const long long dst = ei[(long long)EE + e];
  const float hpv = hp[src * 32 + lane];
  const float hnv = hn[src * 32 + lane];
  if (w > 0.0f) {
    atomAddF(s1 + dst * 32 + lane, hpv);
    atomAddF(s3 + dst * 32 + lane, hnv);
  } else {
    atomAddF(s2 + dst * 32 + lane, hnv);
    atomAddF(s4 + dst * 32 + lane, hpv);
  }
}

// -------- Layer 2: per 16-node tile (one wave):
// out[:, 0:32]  = relu([a1 a2] @ w2_pos_l + hp @ w2_pos_r + b2_pos_r)
// out[:,32:64]  = relu([a3 a4] @ w2_neg_l + hn @ w2_neg_r + b2_neg_r)
// a1 = s1*rp, a2 = s2*rn, a3 = s3*rp, a4 = s4*rn  (fused into A loads)
__global__ void gemm_layer2(const float* __restrict__ hp,
                            const float* __restrict__ hn,
                            const float* __restrict__ s1,
                            const float* __restrict__ s2,
                            const float* __restrict__ s3,
                            const float* __restrict__ s4,
                            const float* __restrict__ cnt_p,
                            const float* __restrict__ cnt_n,
                            const float* __restrict__ w2pl,
                            const float* __restrict__ w2pr,
                            const float* __restrict__ b2p,
                            const float* __restrict__ w2nl,
                            const float* __restrict__ w2nr,
                            const float* __restrict__ b2n,
                            float* __restrict__ out) {
  const int lane  = threadIdx.x & 31;
  const int tile  = blockIdx.x * (blockDim.x >> 5) + (threadIdx.x >> 5);
  if (tile >= TILES) return;
  const int col   = lane & 15;
  const int khalf = lane >> 4;
  const int nodeA = tile * 16 + col;

  const float rp = 1.0f / fmaxf(cnt_p[nodeA], 1.0f);
  const float rn = 1.0f / fmaxf(cnt_n[nodeA], 1.0f);
  const float* A1 = s1 + (size_t)nodeA * 32;
  const float* A2 = s2 + (size_t)nodeA * 32;
  const float* A3 = s3 + (size_t)nodeA * 32;
  const float* A4 = s4 + (size_t)nodeA * 32;
  const float* Ahp = hp + (size_t)nodeA * 32;
  const float* Ahn = hn + (size_t)nodeA * 32;
  const int rowBase = tile * 16 + khalf * 8;

#pragma unroll
  for (int cb = 0; cb < 32; cb += 16) {
    // positive output columns [cb, cb+16)
    v8f c;
    {
      const float bv = b2p[cb + col];
#pragma unroll
      for (int r = 0; r < 8; ++r) c[r] = bv;
    }
    c = wmma_mac<32>(c, A1, rp, w2pl + cb + col, khalf);              // W rows 0..31
    c = wmma_mac<32>(c, A2, rn, w2pl + 32 * 32 + cb + col, khalf);    // W rows 32..63
    c = wmma_mac<32>(c, Ahp, 1.0f, w2pr + cb + col, khalf);
#pragma unroll
    for (int r = 0; r < 8; ++r)
      out[(size_t)(rowBase + r) * 64 + cb + col] = fmaxf(c[r], 0.0f);

    // negative output columns [32+cb, 32+cb+16)
    v8f d;
    {
      const float bv = b2n[cb + col];
#pragma unroll
      for (int r = 0; r < 8; ++r) d[r] = bv;
    }
    d = wmma_mac<32>(d, A3, rp, w2nl + cb + col, khalf);
    d = wmma_mac<32>(d, A4, rn, w2nl + 32 * 32 + cb + col, khalf);
    d = wmma_mac<32>(d, Ahn, 1.0f, w2nr + cb + col, khalf);
#pragma unroll
    for (int r = 0; r < 8; ++r)
      out[(size_t)(rowBase + r) * 64 + 32 + cb + col] = fmaxf(d[r], 0.0f);
  }
}

extern "C" void kernel_launch(void* const* d_in, const int* in_sizes, int n_in,
                              void* d_out, int out_size, void* d_ws, size_t ws_size,
                              hipStream_t stream) {
  const float*     x    = (const float*)d_in[0];
  const long long* ei   = (const long long*)d_in[1];
  const float*     ew   = (const float*)d_in[2];
  const float*     w1pl = (const float*)d_in[3];
  const float*     w1pr = (const float*)d_in[4];
  const float*     b1p  = (const float*)d_in[5];
  const float*     w1nl = (const float*)d_in[6];
  const float*     w1nr = (const float*)d_in[7];
  const float*     b1n  = (const float*)d_in[8];
  const float*     w2pl = (const float*)d_in[9];
  const float*     w2pr = (const float*)d_in[10];
  const float*     b2p  = (const float*)d_in[11];
  const float*     w2nl = (const float*)d_in[12];
  const float*     w2nr = (const float*)d_in[13];
  const float*     b2n  = (const float*)d_in[14];
  float*           out  = (float*)d_out;

  // Workspace layout (floats). Total = N*(64+64+1+1+32+32)*4 B = 77.6 MB.
  float* ws    = (float*)d_ws;
  float* sum_p = ws;                              // N*64
  float* sum_n = ws + (size_t)NN * 64;            // N*64
  float* cnt_p = ws + (size_t)NN * 128;           // N
  float* cnt_n = cnt_p + NN;                      // N
  float* hp    = cnt_n + NN;                      // N*32
  float* hn    = hp + (size_t)NN * 32;            // N*32
  // Layer-2 accumulators alias the (consumed) layer-1 sums:
  float* s1 = sum_p;
  float* s2 = sum_p + (size_t)NN * 32;
  float* s3 = sum_n;
  float* s4 = sum_n + (size_t)NN * 32;

  const int edgeBlocks = EE / 8;                  // 8 wave32 per 256-thread block
  const int gemmBlocks = (TILES + 3) / 4;         // 4 waves (tiles) per block

  // sums + counts are contiguous: zero in one memset
  hipMemsetAsync(sum_p, 0, (size_t)(NN * 128 + 2 * NN) * sizeof(float), stream);
  edge_pass1<<<edgeBlocks, 256, 0, stream>>>(ei, ew, x, sum_p, sum_n, cnt_p, cnt_n);
  gemm_layer1<<<gemmBlocks, 128, 0, stream>>>(x, sum_p, sum_n, cnt_p, cnt_n,
                                              w1pl, w1pr, b1p, w1nl, w1nr, b1n,
                                              hp, hn);
  hipMemsetAsync(sum_p, 0, (size_t)NN * 128 * sizeof(float), stream);
  edge_pass2<<<edgeBlocks, 256, 0, stream>>>(ei, ew, hp, hn, s1, s2, s3, s4);
  gemm_layer2<<<gemmBlocks, 128, 0, stream>>>(hp, hn, s1, s2, s3, s4,
                                              cnt_p, cnt_n,
                                              w2pl, w2pr, b2p, w2nl, w2nr, b2n,
                                              out);
}